// MPNNFlockingModel_75943611728683
// MI455X (gfx1250) — compile-verified
//
#include <hip/hip_runtime.h>
#include <hip/hip_bf16.h>

#define EMBD 4
#define EPSV 1e-5f

typedef __attribute__((ext_vector_type(2))) float v2f;
typedef __attribute__((ext_vector_type(8))) float v8f;

// D = A(16x4 f32) * B(4x16 f32) + C(16x16 f32), wave32 WMMA
__device__ __forceinline__ v8f wmma4(v2f a, v2f b, v8f c) {
    return __builtin_amdgcn_wmma_f32_16x16x4_f32(
        /*neg_a=*/false, a, /*neg_b=*/false, b,
        /*c_mod=*/(short)0, c, /*reuse_a=*/false, /*reuse_b=*/false);
}

// B-matrix lane layout (mirror of documented 32-bit A layout):
// lane holds b[j] = W[K = j + 2*(lane>>4)][N = lane&15]; columns >= EMBD padded 0.
__device__ __forceinline__ v2f loadB(const float* __restrict__ Wrow0, int n, int kh) {
    v2f b; b.x = 0.f; b.y = 0.f;
    if (n < EMBD) {
        b.x = Wrow0[(2 * kh + 0) * EMBD + n];
        b.y = Wrow0[(2 * kh + 1) * EMBD + n];
    }
    return b;
}

__device__ __forceinline__ v8f splatC(float bv) {
    v8f c;
#pragma unroll
    for (int j = 0; j < 8; ++j) c[j] = bv;
    return c;
}

// Training-mode BN: scale/shift per column from (sum, sumsq) stats.
__device__ __forceinline__ void bn_params(const float* __restrict__ st,
                                          const float* __restrict__ g,
                                          const float* __restrict__ be,
                                          float inv_cnt, int n,
                                          float* scale, float* shift) {
    int c = (n < EMBD) ? n : 0;
    float m  = st[c] * inv_cnt;
    float v  = st[EMBD + c] * inv_cnt - m * m;
    float sc = g[c] * rsqrtf(v + EPSV);
    *scale = (n < EMBD) ? sc : 0.f;
    *shift = (n < EMBD) ? (be[c] - m * sc) : 0.f;
}

// Stats accumulate: fast path has no per-element masks (padded cols are exactly 0).
__device__ __forceinline__ void acc_stats(const v8f& x, int base, int kh, int n,
                                          int cnt, bool full,
                                          float* acc_s, float* acc_s2) {
    if (full) {
#pragma unroll
        for (int j = 0; j < 8; ++j) {
            float v = x[j];
            *acc_s  += v;
            *acc_s2 += v * v;
        }
    } else if (n < EMBD) {
#pragma unroll
        for (int j = 0; j < 8; ++j) {
            int em = base + j + 8 * kh;     // D element: M = j + 8*kh, N = n
            if (em < cnt) { float v = x[j]; *acc_s += v; *acc_s2 += v * v; }
        }
    }
}

__global__ void init_kernel(const float* __restrict__ pos, const float* __restrict__ vel,
                            float* __restrict__ h, float* __restrict__ aggr,
                            float* __restrict__ stats, int N) {
    int i = blockIdx.x * blockDim.x + threadIdx.x;
    if (i < 32) stats[i] = 0.f;
    if (i < N) {
        h[4 * i + 0] = pos[2 * i + 0];
        h[4 * i + 1] = pos[2 * i + 1];
        h[4 * i + 2] = vel[2 * i + 0];
        h[4 * i + 3] = vel[2 * i + 1];
        aggr[4 * i + 0] = 0.f; aggr[4 * i + 1] = 0.f;
        aggr[4 * i + 2] = 0.f; aggr[4 * i + 3] = 0.f;
    }
}

// MODE 0: x1 stats; MODE 1: bn1->x2 stats; MODE 2: bn2->msg scatter-add to aggr.
template <int MODE>
__global__ void msg_pass(const float* __restrict__ h,
                         const int* __restrict__ srcI, const int* __restrict__ dstI,
                         const float* __restrict__ W1, const float* __restrict__ b1,
                         const float* __restrict__ g1, const float* __restrict__ be1,
                         const float* __restrict__ W2, const float* __restrict__ b2,
                         const float* __restrict__ g2, const float* __restrict__ be2,
                         const float* __restrict__ st1, const float* __restrict__ st2,
                         float* __restrict__ stats_out, float* __restrict__ aggr,
                         int E, int n_tiles) {
    __shared__ float s_y[8 * 64];       // per-wave 16x4 transpose tile
    __shared__ float s_st[2 * EMBD];

    const int tid  = threadIdx.x;
    const int lane = tid & 31;
    const int wave = tid >> 5;
    const int n    = lane & 15;
    const int kh   = lane >> 4;
    float* my = s_y + wave * 64;

    const v2f B1a = loadB(W1,            n, kh);   // K = 0..3  (h_i feats)
    const v2f B1b = loadB(W1 + 4 * EMBD, n, kh);   // K = 4..7  (h_j feats)
    const v2f B2  = loadB(W2,            n, kh);
    const float bv1 = (n < EMBD) ? b1[n] : 0.f;    // hoisted bias loads
    const float bv2 = (n < EMBD) ? b2[n] : 0.f;

    float sc1 = 0.f, sh1 = 0.f, sc2 = 0.f, sh2 = 0.f;
    const float inv_cnt = 1.0f / (float)E;
    if (MODE >= 1) bn_params(st1, g1, be1, inv_cnt, n, &sc1, &sh1);
    if (MODE == 2) bn_params(st2, g2, be2, inv_cnt, n, &sc2, &sh2);

    float acc_s = 0.f, acc_s2 = 0.f;

    for (int tile = blockIdx.x * 8 + wave; tile < n_tiles; tile += gridDim.x * 8) {
        const int base = tile * 16;
        const bool full = (base + 16 <= E);
        const int e  = base + n;
        const int ec = (e < E) ? e : (E - 1);
        const int di = dstI[ec];
        const int si = srcI[ec];

        { // prefetch next tile's indices into L2/L0
            int nt = tile + gridDim.x * 8;
            if (nt < n_tiles) {
                __builtin_prefetch(&dstI[nt * 16 + n], 0, 1);
                __builtin_prefetch(&srcI[nt * 16 + n], 0, 1);
            }
        }

        // A layout: a[j] = feat[K = j + 2*kh] of row M = n
        v2f a0 = *(const v2f*)(h + (size_t)di * 4 + 2 * kh);
        v2f a1 = *(const v2f*)(h + (size_t)si * 4 + 2 * kh);

        v8f x1 = splatC(bv1);
        x1 = wmma4(a0, B1a, x1);
        x1 = wmma4(a1, B1b, x1);

        if (MODE == 0) {
            acc_stats(x1, base, kh, n, E, full, &acc_s, &acc_s2);
            continue;
        }

        // BN1 + ReLU, transpose D-layout -> A-layout via LDS
        if (n < EMBD) {
#pragma unroll
            for (int j = 0; j < 8; ++j)
                my[(j + 8 * kh) * EMBD + n] = fmaxf(x1[j] * sc1 + sh1, 0.f);
        }
        __builtin_amdgcn_wave_barrier();
        v2f a2;
        a2.x = my[n * EMBD + (2 * kh + 0)];
        a2.y = my[n * EMBD + (2 * kh + 1)];
        __builtin_amdgcn_wave_barrier();

        v8f x2 = splatC(bv2);
        x2 = wmma4(a2, B2, x2);

        if (MODE == 1) {
            acc_stats(x2, base, kh, n, E, full, &acc_s, &acc_s2);
            continue;
        }

        // MODE 2: BN2 + ReLU -> msg; scatter-add to aggr[dst]
        if (n < EMBD) {
#pragma unroll
            for (int j = 0; j < 8; ++j)
                my[(j + 8 * kh) * EMBD + n] = fmaxf(x2[j] * sc2 + sh2, 0.f);
        }
        __builtin_amdgcn_wave_barrier();
        if (lane < 16 && e < E) {
#pragma unroll
            for (int c = 0; c < EMBD; ++c)
                atomicAdd(&aggr[(size_t)di * EMBD + c], my[lane * EMBD + c]);
        }
        __builtin_amdgcn_wave_barrier();
    }

    if (MODE == 0 || MODE == 1) {
        if (tid < 2 * EMBD) s_st[tid] = 0.f;
        __syncthreads();
        if (n < EMBD) {
            atomicAdd(&s_st[n], acc_s);          // ds_add_f32
            atomicAdd(&s_st[EMBD + n], acc_s2);
        }
        __syncthreads();
        if (tid < 2 * EMBD) atomicAdd(&stats_out[tid], s_st[tid]);
    }
}

// Same 3-pass scheme over nodes; input row = [h[i] | aggr[i]]; MODE 2 fuses pred head.
template <int MODE>
__global__ void upd_pass(const float* __restrict__ h, const float* __restrict__ aggr,
                         const float* __restrict__ W1, const float* __restrict__ b1,
                         const float* __restrict__ g1, const float* __restrict__ be1,
                         const float* __restrict__ W2, const float* __restrict__ b2,
                         const float* __restrict__ g2, const float* __restrict__ be2,
                         const float* __restrict__ st1, const float* __restrict__ st2,
                         float* __restrict__ stats_out,
                         const float* __restrict__ predW, const float* __restrict__ predB,
                         float* __restrict__ out,
                         int N, int n_tiles) {
    __shared__ float s_y[8 * 64];
    __shared__ float s_st[2 * EMBD];

    const int tid  = threadIdx.x;
    const int lane = tid & 31;
    const int wave = tid >> 5;
    const int n    = lane & 15;
    const int kh   = lane >> 4;
    float* my = s_y + wave * 64;

    const v2f B1a = loadB(W1,            n, kh);
    const v2f B1b = loadB(W1 + 4 * EMBD, n, kh);
    const v2f B2  = loadB(W2,            n, kh);
    const float bv1 = (n < EMBD) ? b1[n] : 0.f;
    const float bv2 = (n < EMBD) ? b2[n] : 0.f;

    // hoisted pred head constants (uniform -> scalar loads)
    float pw[8] = {0}, pb0 = 0.f, pb1 = 0.f;
    if (MODE == 2) {
#pragma unroll
        for (int k = 0; k < 8; ++k) pw[k] = predW[k];
        pb0 = predB[0]; pb1 = predB[1];
    }

    float sc1 = 0.f, sh1 = 0.f, sc2 = 0.f, sh2 = 0.f;
    const float inv_cnt = 1.0f / (float)N;
    if (MODE >= 1) bn_params(st1, g1, be1, inv_cnt, n, &sc1, &sh1);
    if (MODE == 2) bn_params(st2, g2, be2, inv_cnt, n, &sc2, &sh2);

    float acc_s = 0.f, acc_s2 = 0.f;

    for (int tile = blockIdx.x * 8 + wave; tile < n_tiles; tile += gridDim.x * 8) {
        const int base = tile * 16;
        const bool full = (base + 16 <= N);
        const int r  = base + n;
        const int rc = (r < N) ? r : (N - 1);

        v2f a0 = *(const v2f*)(h    + (size_t)rc * 4 + 2 * kh);  // feats 0..3
        v2f a1 = *(const v2f*)(aggr + (size_t)rc * 4 + 2 * kh);  // feats 4..7

        v8f x1 = splatC(bv1);
        x1 = wmma4(a0, B1a, x1);
        x1 = wmma4(a1, B1b, x1);

        if (MODE == 0) {
            acc_stats(x1, base, kh, n, N, full, &acc_s, &acc_s2);
            continue;
        }

        if (n < EMBD) {
#pragma unroll
            for (int j = 0; j < 8; ++j)
                my[(j + 8 * kh) * EMBD + n] = fmaxf(x1[j] * sc1 + sh1, 0.f);
        }
        __builtin_amdgcn_wave_barrier();
        v2f a2;
        a2.x = my[n * EMBD + (2 * kh + 0)];
        a2.y = my[n * EMBD + (2 * kh + 1)];
        __builtin_amdgcn_wave_barrier();

        v8f x2 = splatC(bv2);
        x2 = wmma4(a2, B2, x2);

        if (MODE == 1) {
            acc_stats(x2, base, kh, n, N, full, &acc_s, &acc_s2);
            continue;
        }

        // MODE 2: BN2 + ReLU -> upd; pred head (4 -> 2) and store
        if (n < EMBD) {
#pragma unroll
            for (int j = 0; j < 8; ++j)
                my[(j + 8 * kh) * EMBD + n] = fmaxf(x2[j] * sc2 + sh2, 0.f);
        }
        __builtin_amdgcn_wave_barrier();
        if (lane < 16 && r < N) {
            float u0 = my[lane * EMBD + 0], u1 = my[lane * EMBD + 1];
            float u2 = my[lane * EMBD + 2], u3 = my[lane * EMBD + 3];
            // predW is [EMBD][2] row-major
            float o0 = u0 * pw[0] + u1 * pw[2] + u2 * pw[4] + u3 * pw[6] + pb0;
            float o1 = u0 * pw[1] + u1 * pw[3] + u2 * pw[5] + u3 * pw[7] + pb1;
            out[(size_t)r * 2 + 0] = o0;
            out[(size_t)r * 2 + 1] = o1;
        }
        __builtin_amdgcn_wave_barrier();
    }

    if (MODE == 0 || MODE == 1) {
        if (tid < 2 * EMBD) s_st[tid] = 0.f;
        __syncthreads();
        if (n < EMBD) {
            atomicAdd(&s_st[n], acc_s);
            atomicAdd(&s_st[EMBD + n], acc_s2);
        }
        __syncthreads();
        if (tid < 2 * EMBD) atomicAdd(&stats_out[tid], s_st[tid]);
    }
}

extern "C" void kernel_launch(void* const* d_in, const int* in_sizes, int n_in,
                              void* d_out, int out_size, void* d_ws, size_t ws_size,
                              hipStream_t stream) {
    const float* pos     = (const float*)d_in[0];
    const float* vel     = (const float*)d_in[1];
    const int*   eidx    = (const int*)d_in[2];
    const float* msg_W1  = (const float*)d_in[3];
    const float* msg_b1  = (const float*)d_in[4];
    const float* msg_g1  = (const float*)d_in[5];
    const float* msg_be1 = (const float*)d_in[6];
    const float* msg_W2  = (const float*)d_in[7];
    const float* msg_b2  = (const float*)d_in[8];
    const float* msg_g2  = (const float*)d_in[9];
    const float* msg_be2 = (const float*)d_in[10];
    const float* upd_W1  = (const float*)d_in[11];
    const float* upd_b1  = (const float*)d_in[12];
    const float* upd_g1  = (const float*)d_in[13];
    const float* upd_be1 = (const float*)d_in[14];
    const float* upd_W2  = (const float*)d_in[15];
    const float* upd_b2  = (const float*)d_in[16];
    const float* upd_g2  = (const float*)d_in[17];
    const float* upd_be2 = (const float*)d_in[18];
    const float* pred_W  = (const float*)d_in[19];
    const float* pred_b  = (const float*)d_in[20];

    const int N = in_sizes[0] / 2;
    const int E = in_sizes[2] / 2;
    const int* srcI = eidx;        // edge_index[0]
    const int* dstI = eidx + E;    // edge_index[1]

    float* ws    = (float*)d_ws;
    float* h     = ws;                       // [N*4]
    float* aggr  = ws + (size_t)4 * N;       // [N*4]
    float* st    = ws + (size_t)8 * N;       // [32]: msg1, msg2, upd1, upd2
    float* st_m1 = st +  0;
    float* st_m2 = st +  8;
    float* st_u1 = st + 16;
    float* st_u2 = st + 24;

    dim3 blk(256);
    int gInit = (N + 255) / 256;
    init_kernel<<<gInit, blk, 0, stream>>>(pos, vel, h, aggr, st, N);

    const int tilesE = (E + 15) / 16;
    int gE = (tilesE + 7) / 8; if (gE > 4096) gE = 4096;
    msg_pass<0><<<gE, blk, 0, stream>>>(h, srcI, dstI,
        msg_W1, msg_b1, msg_g1, msg_be1, msg_W2, msg_b2, msg_g2, msg_be2,
        nullptr, nullptr, st_m1, nullptr, E, tilesE);
    msg_pass<1><<<gE, blk, 0, stream>>>(h, srcI, dstI,
        msg_W1, msg_b1, msg_g1, msg_be1, msg_W2, msg_b2, msg_g2, msg_be2,
        st_m1, nullptr, st_m2, nullptr, E, tilesE);
    msg_pass<2><<<gE, blk, 0, stream>>>(h, srcI, dstI,
        msg_W1, msg_b1, msg_g1, msg_be1, msg_W2, msg_b2, msg_g2, msg_be2,
        st_m1, st_m2, nullptr, aggr, E, tilesE);

    const int tilesN = (N + 15) / 16;
    int gN = (tilesN + 7) / 8; if (gN > 2048) gN = 2048;
    upd_pass<0><<<gN, blk, 0, stream>>>(h, aggr,
        upd_W1, upd_b1, upd_g1, upd_be1, upd_W2, upd_b2, upd_g2, upd_be2,
        nullptr, nullptr, st_u1, nullptr, nullptr, nullptr, N, tilesN);
    upd_pass<1><<<gN, blk, 0, stream>>>(h, aggr,
        upd_W1, upd_b1, upd_g1, upd_be1, upd_W2, upd_b2, upd_g2, upd_be2,
        st_u1, nullptr, st_u2, nullptr, nullptr, nullptr, N, tilesN);
    upd_pass<2><<<gN, blk, 0, stream>>>(h, aggr,
        upd_W1, upd_b1, upd_g1, upd_be1, upd_W2, upd_b2, upd_g2, upd_be2,
        st_u1, st_u2, nullptr, pred_W, pred_b, (float*)d_out, N, tilesN);
}